// InhibitionMemoryLayer_37357625541084
// MI455X (gfx1250) — compile-verified
//
#include <hip/hip_runtime.h>
#include <hip/hip_bf16.h>
#include <math.h>

// Sizes from the reference
#define HID   512
#define INH   512
#define INSZ  256
#define BATCH 4
#define SEQ   64
#define DELAY 8
#define LR_F    0.01f
#define DECAY_F 0.951229424500714f     // exp(-1/20)
#define OMD_F   0.048770575499285991f  // 1 - DECAY

typedef float v2f __attribute__((ext_vector_type(2)));
typedef float v8f __attribute__((ext_vector_type(8)));

// ---------------------------------------------------------------------------
// Kernel A: input projection  i[m,o] = sum_c x[m,c] * W[o,c]
//   M = B*T = 256, N = 2H = 1024, K = 256.  f32 WMMA 16x16x4.
// A layout (16x4 f32): lanes 0-15 -> M=lane, v0=K0,v1=K1; lanes 16-31 -> K2,K3.
// B layout (4x16):     lanes 0-15 -> N=lane, v0=K0,v1=K1; lanes 16-31 -> K2,K3.
// ---------------------------------------------------------------------------
__global__ __launch_bounds__(256) void k_proj(const float* __restrict__ x,
                                              const float* __restrict__ W,
                                              float* __restrict__ iproj) {
  const int gw   = (blockIdx.x * 256 + threadIdx.x) >> 5;   // global wave id
  const int lane = threadIdx.x & 31;
  const int tm = gw >> 6;          // 16 tiles along M
  const int tn = gw & 63;          // 64 tiles along N
  const int hf = lane >> 4, ln = lane & 15;
  const float* ar = x + (size_t)(tm * 16 + ln) * INSZ;
  const float* br = W + (size_t)(tn * 16 + ln) * INSZ;
  v8f acc = {};
  for (int c = 0; c < INSZ; c += 4) {
    const int cc = c + 2 * hf;
    v2f a, b;
    a.x = ar[cc]; a.y = ar[cc + 1];
    b.x = br[cc]; b.y = br[cc + 1];
    acc = __builtin_amdgcn_wmma_f32_16x16x4_f32(false, a, false, b,
                                                (short)0, acc, false, false);
  }
  for (int r = 0; r < 8; ++r)
    iproj[(size_t)(tm * 16 + r + 8 * hf) * 1024 + tn * 16 + ln] = acc[r];
}

// ---------------------------------------------------------------------------
// Kernel B: key neuron chain (depends only on ik) — precompute all key_o.
// ---------------------------------------------------------------------------
__global__ __launch_bounds__(512) void k_keys(const float* __restrict__ iproj,
                                              float* __restrict__ keyH,
                                              float* __restrict__ keys_out) {
  const int b = blockIdx.x, h = threadIdx.x;
  float kv = 0.f;
  for (int t = 0; t < SEQ; ++t) {
    kv = DECAY_F * kv + iproj[((size_t)b * SEQ + t) * 1024 + h];
    const float ko = tanhf(kv);
    keyH[((size_t)b * SEQ + t) * HID + h]     = ko;
    keys_out[((size_t)b * SEQ + t) * HID + h] = ko;
  }
}

// ---------------------------------------------------------------------------
// Kernel C: ikv0[b,t,h] = sum_k mem0[b,h,k] * key[b,t,k]   (WMMA, per-batch
// GEMM M=512(h), N=64(t), K=512).  0.2 scale applied in the scan.
// ---------------------------------------------------------------------------
__global__ __launch_bounds__(256) void k_ikv0(const float* __restrict__ mem0,
                                              const float* __restrict__ keyH,
                                              float* __restrict__ ikv0) {
  const int gw   = (blockIdx.x * 256 + threadIdx.x) >> 5;
  const int lane = threadIdx.x & 31;
  const int b  = gw >> 7;          // 128 tiles per batch
  const int rr = gw & 127;
  const int tm = rr >> 2;          // 32 tiles along h
  const int tn = rr & 3;           // 4 tiles along t
  const int hf = lane >> 4, ln = lane & 15;
  const float* ar = mem0 + ((size_t)b * HID + tm * 16 + ln) * HID;
  const float* br = keyH + ((size_t)b * SEQ + tn * 16 + ln) * HID;
  v8f acc = {};
  for (int c = 0; c < HID; c += 4) {
    const int cc = c + 2 * hf;
    v2f a, bb;
    a.x  = ar[cc]; a.y  = ar[cc + 1];
    bb.x = br[cc]; bb.y = br[cc + 1];
    acc = __builtin_amdgcn_wmma_f32_16x16x4_f32(false, a, false, bb,
                                                (short)0, acc, false, false);
  }
  for (int r = 0; r < 8; ++r) {
    const int hh = tm * 16 + r + 8 * hf;
    const int tt = tn * 16 + ln;
    ikv0[((size_t)b * SEQ + tt) * HID + hh] = acc[r];
  }
}

// ---------------------------------------------------------------------------
// Kernel D: the sequential scan. One block per batch, 512 threads (thread h
// owns row/element h). Low-rank form:
//   ikv_t[h]    = 0.2*(ikv0[t,h] + LR * sum_{s<t} ktr_s[h] * (vtr_s . key_t))
//   inh_out[j]  = (offdiag(ie0) . buf_t)[j] + LR*sum_{s<t} itr_s[j]*(vtr_s.buf_t)
//                 - LR*accDiag[j]*buf_t[j]
//   inh_in[k]   = val_o[k] * diag_ei[k]
// vtr history lives in LDS (dot products); ktr/itr histories go to global ws
// and are only ever re-read by the thread that wrote them (plus later kernels).
// ---------------------------------------------------------------------------
__global__ __launch_bounds__(512) void k_scan(const float* __restrict__ iproj,
                                              const float* __restrict__ keyH,
                                              const float* __restrict__ ikv0,
                                              const float* __restrict__ ei0,
                                              const float* __restrict__ ie0,
                                              float* __restrict__ ktrH,
                                              float* __restrict__ vtrH,
                                              float* __restrict__ itrH,
                                              float* __restrict__ dei,
                                              float* __restrict__ vals_out) {
  __shared__ __align__(16) float vtr_l[SEQ][HID];   // 128 KB
  __shared__ __align__(16) float ring[DELAY][HID];  // 16 KB delay buffer
  __shared__ __align__(16) float key_s[HID];
  __shared__ float dk[SEQ];
  __shared__ float db[SEQ];

  const int b = blockIdx.x;
  const int h = threadIdx.x;
  const int lane = h & 31, wv = h >> 5;

  for (int r = 0; r < DELAY; ++r) ring[r][h] = 0.f;

  float vvs = 0.f, isv = 0.f;
  float ktr = 0.f, vtr = 0.f, itr = 0.f, accD = 0.f;
  const size_t mbase = (size_t)b * HID * HID + (size_t)h * HID;
  float d_ei = ei0[mbase + h];                 // diag of masked ei
  const float ie0d = ie0[mbase + h];           // ie0 diagonal element
  const float4* ierow = (const float4*)(ie0 + mbase);
  __syncthreads();

  for (int t = 0; t < SEQ; ++t) {
    const int r = t & (DELAY - 1);
    const size_t rowoff = ((size_t)b * SEQ + t) * HID;

    key_s[h] = keyH[rowoff + h];
    __syncthreads();

    // dot products of vtr history with key_t and buf_t (waves split the s's)
    for (int s = wv; s < t; s += 16) {
      float p1 = 0.f, p2 = 0.f;
      for (int k = lane; k < HID; k += 32) {
        const float vk = vtr_l[s][k];
        p1 = fmaf(vk, key_s[k], p1);
        p2 = fmaf(vk, ring[r][k], p2);
      }
      for (int off = 16; off > 0; off >>= 1) {
        p1 += __shfl_xor(p1, off, 32);
        p2 += __shfl_xor(p2, off, 32);
      }
      if (lane == 0) { dk[s] = p1; db[s] = p2; }
    }
    __syncthreads();

    // history-weighted sums (own column h; self-written global, L2-resident)
    float sk = 0.f, si = 0.f;
    for (int s = 0; s < t; ++s) {
      sk = fmaf(ktrH[((size_t)b * SEQ + s) * HID + h], dk[s], sk);
      si = fmaf(itrH[((size_t)b * SEQ + s) * HID + h], db[s], si);
    }

    // ie0 row matvec against buf_t (float4, L2-resident)
    float mv = 0.f;
    const float4* b4 = (const float4*)ring[r];
    for (int k4 = 0; k4 < HID / 4; ++k4) {
      if ((k4 & 15) == 0 && k4 + 16 < HID / 4)
        __builtin_prefetch(&ierow[k4 + 16], 0, 3);   // global_prefetch_b8
      const float4 a = ierow[k4], c = b4[k4];
      mv = fmaf(a.x, c.x, mv); mv = fmaf(a.y, c.y, mv);
      mv = fmaf(a.z, c.z, mv); mv = fmaf(a.w, c.w, mv);
    }
    const float bufh = ring[r][h];
    const float inh_out = mv + LR_F * si - bufh * (ie0d + LR_F * accD);

    const float ikv = 0.2f * (ikv0[rowoff + h] + LR_F * sk);
    const float ivt = iproj[((size_t)b * SEQ + t) * 1024 + HID + h];

    vvs = DECAY_F * vvs + ivt + ikv - inh_out;
    const float val = tanhf(vvs);
    isv = DECAY_F * isv + val * d_ei;
    const float inh = tanhf(isv);

    ktr = DECAY_F * ktr + OMD_F * key_s[h];
    vtr = DECAY_F * vtr + OMD_F * val;
    itr = DECAY_F * itr + OMD_F * inh;
    const float p = vtr * itr;
    d_ei += LR_F * p;
    accD += p;

    ktrH[rowoff + h] = ktr;
    vtrH[rowoff + h] = vtr;
    itrH[rowoff + h] = itr;
    vtr_l[t][h] = vtr;
    vals_out[rowoff + h] = val;

    __syncthreads();          // all reads of ring[r] done
    ring[r][h] = inh;         // becomes buf_t for step t+DELAY
  }
  dei[(size_t)b * HID + h] = d_ei;
}

// ---------------------------------------------------------------------------
// Kernel E (WMMA): materialize mem_f and ie_f.
//   mem_f = mem0 + LR * sum_{s=0..63} ktr_s (x) vtr_s
//   ie_f[j,k] = (j!=k) ? ie0[j,k] + LR*S63[j,k] : LR*itr63[j]*vtr63[j]
// Per-batch GEMM M=512, N=512, K=64.  blockIdx.y selects matrix.
// ---------------------------------------------------------------------------
__global__ __launch_bounds__(256) void k_final(const float* __restrict__ mem0,
                                               const float* __restrict__ ie0,
                                               const float* __restrict__ ktrH,
                                               const float* __restrict__ vtrH,
                                               const float* __restrict__ itrH,
                                               float* __restrict__ out) {
  const int which = blockIdx.y;  // 0 = mem, 1 = ie
  const int gw    = (blockIdx.x * 256 + threadIdx.x) >> 5;
  const int lane  = threadIdx.x & 31;
  const int b  = gw >> 10;       // 1024 tiles per batch
  const int rr = gw & 1023;
  const int tm = rr >> 5, tn = rr & 31;
  const int hf = lane >> 4, ln = lane & 15;
  const float* Ah = which ? itrH : ktrH;
  v8f acc = {};
  for (int c = 0; c < SEQ; c += 4) {
    const int cc = c + 2 * hf;
    v2f a, bb;
    a.x  = Ah[((size_t)b * SEQ + cc)     * HID + tm * 16 + ln];
    a.y  = Ah[((size_t)b * SEQ + cc + 1) * HID + tm * 16 + ln];
    bb.x = vtrH[((size_t)b * SEQ + cc)     * HID + tn * 16 + ln];
    bb.y = vtrH[((size_t)b * SEQ + cc + 1) * HID + tn * 16 + ln];
    acc = __builtin_amdgcn_wmma_f32_16x16x4_f32(false, a, false, bb,
                                                (short)0, acc, false, false);
  }
  const float* base0 = which ? ie0 : mem0;
  float* dst = out + (which ? (size_t)2097152 : (size_t)0);  // ie region @ 2*B*H*H/4... = 2,097,152
  for (int r = 0; r < 8; ++r) {
    const int m = tm * 16 + r + 8 * hf;
    const int n = tn * 16 + ln;
    const size_t idx = (size_t)b * HID * HID + (size_t)m * HID + n;
    float v = base0[idx] + LR_F * acc[r];
    if (which && m == n)
      v = LR_F * itrH[((size_t)b * SEQ + 63) * HID + m] *
                 vtrH[((size_t)b * SEQ + 63) * HID + n];
    dst[idx] = v;
  }
}

// ---------------------------------------------------------------------------
// Kernel F: ei_f[b,h,i] = (h==i) ? dei[b,h] : LR * vtr63[b,h] * itr63[b,i]
// ---------------------------------------------------------------------------
__global__ __launch_bounds__(256) void k_ei(const float* __restrict__ dei,
                                            const float* __restrict__ vtrH,
                                            const float* __restrict__ itrH,
                                            float* __restrict__ ei_out) {
  const int idx = blockIdx.x * 256 + threadIdx.x;  // 4*512*512 elements
  const int b = idx >> 18;
  const int h = (idx >> 9) & 511;
  const int i = idx & 511;
  float v;
  if (h == i) v = dei[b * HID + h];
  else        v = LR_F * vtrH[((size_t)b * SEQ + 63) * HID + h] *
                         itrH[((size_t)b * SEQ + 63) * HID + i];
  ei_out[idx] = v;
}

// ---------------------------------------------------------------------------
extern "C" void kernel_launch(void* const* d_in, const int* in_sizes, int n_in,
                              void* d_out, int out_size, void* d_ws, size_t ws_size,
                              hipStream_t stream) {
  (void)in_sizes; (void)n_in; (void)out_size; (void)ws_size;
  const float* x    = (const float*)d_in[0];   // [4,64,256]
  const float* W    = (const float*)d_in[1];   // [1024,256]
  const float* mem0 = (const float*)d_in[2];   // [4,512,512]
  const float* ei0  = (const float*)d_in[3];   // [4,512,512]
  const float* ie0  = (const float*)d_in[4];   // [4,512,512]

  float* out = (float*)d_out;
  float* ei_out   = out + 1048576;             // 1,048,576
  float* keys_out = out + 3145728;             // 131,072
  float* vals_out = out + 3276800;             // 131,072

  float* ws    = (float*)d_ws;                 // ~3.5 MB used
  float* iproj = ws;                           // 262144
  float* keyH  = ws + 262144;                  // 131072
  float* ikv0  = ws + 393216;                  // 131072
  float* ktrH  = ws + 524288;                  // 131072
  float* vtrH  = ws + 655360;                  // 131072
  float* itrH  = ws + 786432;                  // 131072
  float* dei   = ws + 917504;                  // 2048

  k_proj<<<128, 256, 0, stream>>>(x, W, iproj);
  k_keys<<<4, 512, 0, stream>>>(iproj, keyH, keys_out);
  k_ikv0<<<64, 256, 0, stream>>>(mem0, keyH, ikv0);
  k_scan<<<4, 512, 0, stream>>>(iproj, keyH, ikv0, ei0, ie0,
                                ktrH, vtrH, itrH, dei, vals_out);
  dim3 gf(512, 2);
  k_final<<<gf, 256, 0, stream>>>(mem0, ie0, ktrH, vtrH, itrH, out);
  k_ei<<<4096, 256, 0, stream>>>(dei, vtrH, itrH, ei_out);
}